// BiLSTMCRF_80865644249454
// MI455X (gfx1250) — compile-verified
//
#include <hip/hip_runtime.h>
#include <stdint.h>

// Problem constants (from reference): B=8192, T=512, K=12
#define BATCH  8192
#define TLEN   512
#define NTAG   12
#define SPLIT  4               // lanes per batch row (lane quad), 3 columns each
#define COLS   3               // NTAG / SPLIT
#define TILE   4               // time-steps per async staging tile (double buffered)
#define BLOCK  256             // threads per block -> 64 batch rows per block
#define GROUPS (BLOCK / SPLIT) // 64
#define NTILE  (TLEN / TILE)   // 128

// ---- CDNA5 async-to-LDS path (gfx1250), with graceful fallback -------------
#if defined(__gfx1250__) && __has_builtin(__builtin_amdgcn_global_load_async_to_lds_b128)
#define HAVE_ASYNC 1
#else
#define HAVE_ASYNC 0
#endif

#if defined(__gfx1250__) && __has_builtin(__builtin_amdgcn_s_wait_asynccnt)
#define WAIT_ASYNC(n) __builtin_amdgcn_s_wait_asynccnt(n)
#elif defined(__gfx1250__) && HAVE_ASYNC
#define WAIT_ASYNC(n) asm volatile("s_wait_asynccnt %0" ::"n"(n) : "memory")
#else
#define WAIT_ASYNC(n) do { } while (0)
#endif

#if HAVE_ASYNC
typedef int v4i_ __attribute__((vector_size(16)));
typedef __attribute__((address_space(1))) v4i_* gptr_b128;
typedef __attribute__((address_space(3))) v4i_* lptr_b128;
#endif

__device__ __forceinline__ void async_copy_b128(const float* gsrc, float* ldst) {
#if HAVE_ASYNC
    __builtin_amdgcn_global_load_async_to_lds_b128(
        (gptr_b128)gsrc, (lptr_b128)ldst, 0 /*imm offset*/, 0 /*cpol*/);
#else
    *(float4*)ldst = *(const float4*)gsrc;   // synchronous fallback
#endif
}

__global__ __launch_bounds__(BLOCK) void viterbi_fwd_bwd(
    const float* __restrict__ logits,          // [B, T, K]
    const float* __restrict__ trans,           // [K, K]
    float* __restrict__ scores,                // [B]
    int* __restrict__ paths,                   // [B, T]
    unsigned short* __restrict__ bpws)         // [T-1, B, 4] 3x4-bit backptrs per sublane
{
    // Raw transitions staged once; each lane copies its 36 values to VGPRs.
    __shared__ float sTraw[NTAG * NTAG];
    // Double-buffered emissions: [buf][group][step][12]; 48B/group/step, b128-aligned.
    __shared__ __align__(16) float sEm[2][GROUPS][TILE][NTAG];
    // prev[12] exchange slot per group: 5 float4 (1 pad) to spread LDS banks.
    __shared__ __align__(16) float4 sXv[GROUPS * 5];

    const int tid = threadIdx.x;
    const int g   = tid >> 2;          // group (batch row) within block
    const int sub = tid & 3;           // sublane: owns columns j = sub + 4k, k=0..2
    const int b   = blockIdx.x * GROUPS + g;

    for (int x = tid; x < NTAG * NTAG; x += BLOCK) sTraw[x] = trans[x];
    __syncthreads();                   // only barrier in the kernel

    // Per-lane transition registers: wT[k][i] = trans[i][sub + 4k]  (36 VGPRs)
    float wT[COLS][NTAG];
#pragma unroll
    for (int k = 0; k < COLS; ++k)
#pragma unroll
        for (int i = 0; i < NTAG; ++i)
            wT[k][i] = sTraw[i * NTAG + (sub + 4 * k)];

    // Async staging: the 12 b128 chunks of a group's 192B tile (TILE steps x 48B)
    // are dealt round-robin to the 4 sublanes: chunk c = sub + 4*kk -> step c/3,
    // 16B-part c%3. chunkoff is loop-invariant per lane.
    const float* gbase = logits + (size_t)b * TLEN * NTAG;
    int chunkoff[COLS];
#pragma unroll
    for (int kk = 0; kk < COLS; ++kk) {
        const int c = sub + 4 * kk;
        chunkoff[kk] = (c / 3) * NTAG + (c % 3) * 4;   // floats
    }
    float* emgrp[2] = { &sEm[0][g][0][0], &sEm[1][g][0][0] };

    auto stage = [&](int buf, int tile) {
        const float* src0 = gbase + (size_t)tile * TILE * NTAG;
#pragma unroll
        for (int kk = 0; kk < COLS; ++kk)
            async_copy_b128(src0 + chunkoff[kk], emgrp[buf] + chunkoff[kk]);
    };

    stage(0, 0);

    float p0 = 0.f, p1 = 0.f, p2 = 0.f;   // this lane's 3 trellis entries
    float4* myslot = &sXv[g * 5 + sub];
    const float4* grpbase = &sXv[g * 5];

    for (int tile = 0; tile < NTILE; ++tile) {
        const int buf = tile & 1;
        if (tile + 1 < NTILE) {
            stage(buf ^ 1, tile + 1);
            WAIT_ASYNC(COLS);      // this tile landed; next tile (3 ops) in flight
        } else {
            WAIT_ASYNC(0);
        }

#pragma unroll
        for (int s = 0; s < TILE; ++s) {
            const int t = tile * TILE + s;
            const float* em = emgrp[buf] + s * NTAG;
            const float e0 = em[sub], e1 = em[sub + 4], e2 = em[sub + 8];

            if (t == 0) {
                p0 = e0; p1 = e1; p2 = e2;
            } else {
                // ---- quad-internal prev[12] exchange via LDS (same wave: no barrier)
                *myslot = make_float4(p0, p1, p2, 0.f);
                const float4 q0 = grpbase[0], q1 = grpbase[1],
                             q2 = grpbase[2], q3 = grpbase[3];
                const float pa[NTAG] = { q0.x, q1.x, q2.x, q3.x,
                                         q0.y, q1.y, q2.y, q3.y,
                                         q0.z, q1.z, q2.z, q3.z };
                // ---- max-plus + argmax for this lane's 3 columns
                float cur[COLS];
                int bpk[COLS];
#pragma unroll
                for (int k = 0; k < COLS; ++k) {
                    float best = pa[0] + wT[k][0];
                    int bp = 0;
#pragma unroll
                    for (int i = 1; i < NTAG; ++i) {
                        const float c = pa[i] + wT[k][i];
                        const bool gt = c > best;    // strict: first max wins (argmax)
                        best = gt ? c : best;
                        bp   = gt ? i : bp;
                    }
                    cur[k] = best; bpk[k] = bp;
                }
                p0 = cur[0] + e0; p1 = cur[1] + e1; p2 = cur[2] + e2;
                const unsigned short packed =
                    (unsigned short)(bpk[0] | (bpk[1] << 4) | (bpk[2] << 8));
                // coalesced: quad writes 4 consecutive u16 = one u64 per (t,b)
                bpws[((size_t)(t - 1) * BATCH + b) * SPLIT + sub] = packed;
            }
        }
    }

    // Final exchange -> full prev[12]; sublane 0 finishes score + backtrack.
    *myslot = make_float4(p0, p1, p2, 0.f);
    const float4 q0 = grpbase[0], q1 = grpbase[1], q2 = grpbase[2], q3 = grpbase[3];
    const float pa[NTAG] = { q0.x, q1.x, q2.x, q3.x,
                             q0.y, q1.y, q2.y, q3.y,
                             q0.z, q1.z, q2.z, q3.z };
    if (sub == 0) {
        float best = pa[0];
        int last = 0;
#pragma unroll
        for (int j = 1; j < NTAG; ++j) {
            const bool gt = pa[j] > best;
            best = gt ? pa[j] : best;
            last = gt ? j : last;
        }
        scores[b] = best;

        int* mypath = paths + (size_t)b * TLEN;
        mypath[TLEN - 1] = last;
        int c = last;
        const unsigned long long* w64 = (const unsigned long long*)bpws;
        for (int tt = TLEN - 2; tt >= 0; --tt) {
            const unsigned long long pk = w64[(size_t)tt * BATCH + b];  // coalesced
            // tag c lives in sublane (c&3), nibble (c>>2)
            c = (int)((pk >> (16 * (c & 3) + 4 * (c >> 2))) & 0xFull);
            mypath[tt] = c;
        }
    }
}

extern "C" void kernel_launch(void* const* d_in, const int* in_sizes, int n_in,
                              void* d_out, int out_size, void* d_ws, size_t ws_size,
                              hipStream_t stream) {
    const float* logits = (const float*)d_in[0];   // [8192, 512, 12] f32
    const float* trans  = (const float*)d_in[1];   // [12, 12] f32

    float* scores = (float*)d_out;                 // first B floats
    int*   paths  = (int*)d_out + BATCH;           // then B*T int32

    unsigned short* bpws = (unsigned short*)d_ws;  // (T-1)*B*8 bytes = 33.5 MB

    dim3 grid(BATCH / GROUPS), block(BLOCK);       // 128 blocks x 256 thr = 1024 waves
    hipLaunchKernelGGL(viterbi_fwd_bwd, grid, block, 0, stream,
                       logits, trans, scores, paths, bpws);
}